// GINConvLayer_5592047419415
// MI455X (gfx1250) — compile-verified
//
#include <hip/hip_runtime.h>

// ---------------------------------------------------------------------------
// GIN layer for MI455X (gfx1250, wave32):
//   aggr = segment_sum(x[row], col); out = (1+eps)*x + aggr
//   h = out @ W1^T + b1 ; BN(batch stats) ; ReLU ; y = h @ W2^T + b2
// fp32 WMMA (V_WMMA_F32_16X16X4_F32) for both GEMMs: exact fp32 precision,
// and the layer is L2/atomic bound anyway (x and aggr each 51.2MB -> fit in
// the 192MB L2), so fp32 matrix path is the right choice.
// ---------------------------------------------------------------------------

typedef __attribute__((ext_vector_type(2))) float v2f;
typedef __attribute__((ext_vector_type(8))) float v8f;

#define CH 128
#define LDS_STRIDE 132   // 128 + 4 pad: stride mod 64 banks = 4 -> conflict-free column reads

__device__ __forceinline__ v8f wmma_fp32(v2f a, v2f b, v8f c) {
  // 8 args: (neg_a, A, neg_b, B, c_mod, C, reuse_a, reuse_b)
  return __builtin_amdgcn_wmma_f32_16x16x4_f32(false, a, false, b, (short)0, c,
                                               false, false);
}

// ---------------------------------------------------------------------------
// Kernel 1: buf = (1+eps)*x   (also zero the BN stats accumulator: 256 floats)
// ---------------------------------------------------------------------------
__global__ void k_init(const float* __restrict__ x, const float* __restrict__ eps,
                       float* __restrict__ buf, float* __restrict__ sums, int total4) {
  int i = blockIdx.x * blockDim.x + threadIdx.x;
  if (i < 64) {                       // zero 256 floats of BN sums
    float4 z = {0.f, 0.f, 0.f, 0.f};
    ((float4*)sums)[i] = z;
  }
  if (i < total4) {
    float s = 1.0f + eps[0];
    float4 v = ((const float4*)x)[i];
    v.x *= s; v.y *= s; v.z *= s; v.w *= s;
    ((float4*)buf)[i] = v;
  }
}

// ---------------------------------------------------------------------------
// Kernel 2: edge scatter-add. One wave32 per edge: 32 lanes x float4 = 128 ch.
// fp32 HW atomics (global_atomic_add_f32) into L2-resident aggregation buffer.
// ---------------------------------------------------------------------------
__global__ void k_edge_scatter(const float* __restrict__ x,
                               const long long* __restrict__ ei,
                               float* __restrict__ buf, int E) {
  const int wid  = (blockIdx.x * blockDim.x + threadIdx.x) >> 5;
  const int lane = threadIdx.x & 31;
  const int nw   = (gridDim.x * blockDim.x) >> 5;
  for (int e = wid; e < E; e += nw) {
    const long long r = ei[e];
    const long long c = ei[(long long)E + e];
    if (e + nw < E) {                 // prefetch next gather row (global_prefetch)
      const long long rn = ei[e + nw];
      __builtin_prefetch(&x[rn * CH + lane * 4], 0, 3);
    }
    const float4 v = *(const float4*)&x[r * CH + lane * 4];
    float* dst = &buf[c * CH + lane * 4];
    unsafeAtomicAdd(dst + 0, v.x);
    unsafeAtomicAdd(dst + 1, v.y);
    unsafeAtomicAdd(dst + 2, v.z);
    unsafeAtomicAdd(dst + 3, v.w);
  }
}

// ---------------------------------------------------------------------------
// Kernel 3: Out[16-row tile] = A @ W^T + bias via V_WMMA_F32_16X16X4_F32.
// Block = 256 threads = 8 waves; wave w owns output channels [16w,16w+16).
// A tile staged in padded LDS once per block; K accumulated 4 at a time.
// A layout (16x4 fp32): lane<16 -> K={k,k+1}, lane>=16 -> K={k+2,k+3}, M=lane&15.
// B layout symmetric: N=lane&15, same K-pair split.
// C/D: VGPR v holds M=v (lanes 0-15) / M=v+8 (lanes 16-31), N=lane&15.
// ---------------------------------------------------------------------------
__global__ void k_gemm_bias(const float* __restrict__ A, const float* __restrict__ W,
                            const float* __restrict__ bias, float* __restrict__ Out) {
  __shared__ float As[16 * LDS_STRIDE];
  const int tid = threadIdx.x;
  const long long row0 = (long long)blockIdx.x * 16;

  for (int i = tid * 4; i < 16 * CH; i += 256 * 4) {
    const int r = i >> 7, c = i & 127;
    *(float4*)&As[r * LDS_STRIDE + c] = *(const float4*)&A[(row0 + r) * CH + c];
  }
  __syncthreads();

  const int lane  = tid & 31;
  const int mlane = lane & 15;
  const int kofs  = (lane >> 4) << 1;
  const int n0    = (tid >> 5) * 16;

  const float* arow = &As[mlane * LDS_STRIDE];
  const float* wrow = &W[(size_t)(n0 + mlane) * CH];   // B[k][n] = W[n*CH + k]

  v8f acc = {};
  #pragma unroll 4
  for (int k = 0; k < CH; k += 4) {
    v2f a, b;
    a.x = arow[k + kofs]; a.y = arow[k + kofs + 1];
    b.x = wrow[k + kofs]; b.y = wrow[k + kofs + 1];
    acc = wmma_fp32(a, b, acc);
  }

  const int col = n0 + mlane;
  const float bv = bias[col];
  const long long rbase = row0 + ((lane >> 4) << 3);
  #pragma unroll
  for (int v = 0; v < 8; ++v)
    Out[(rbase + v) * CH + col] = acc[v] + bv;
}

// ---------------------------------------------------------------------------
// Kernel 4: per-channel sum / sum-of-squares over all N rows.
// ---------------------------------------------------------------------------
__global__ void k_bn_stats(const float* __restrict__ H, float* __restrict__ sums, int N) {
  const int tid = blockIdx.x * blockDim.x + threadIdx.x;
  const int c = tid & 127;
  const int rstride = (gridDim.x * blockDim.x) >> 7;
  float s = 0.f, sq = 0.f;
  for (int r = tid >> 7; r < N; r += rstride) {
    const float v = H[(size_t)r * CH + c];
    s += v; sq += v * v;
  }
  unsafeAtomicAdd(&sums[c], s);
  unsafeAtomicAdd(&sums[CH + c], sq);
}

// ---------------------------------------------------------------------------
// Kernel 5: fold BN into per-channel scale/shift: s = g*rsqrt(var+eps),
// t = beta - mean*s. One block, 128 threads.
// ---------------------------------------------------------------------------
__global__ void k_bn_finalize(const float* __restrict__ sums,
                              const float* __restrict__ gamma,
                              const float* __restrict__ beta,
                              float* __restrict__ st, float invN) {
  const int c = threadIdx.x;
  const float mean = sums[c] * invN;
  const float var  = sums[CH + c] * invN - mean * mean;
  const float sc   = gamma[c] * rsqrtf(var + 1e-5f);
  st[c]      = sc;
  st[CH + c] = beta[c] - mean * sc;
}

// ---------------------------------------------------------------------------
// Kernel 6: Out = relu(H*s + t) @ W2^T + b2, WMMA, in-place H==Out safe:
// each block reads only the 16 rows it writes, with __syncthreads between.
// ---------------------------------------------------------------------------
__global__ void k_bn_relu_gemm(const float* __restrict__ H, const float* __restrict__ W,
                               const float* __restrict__ bias,
                               const float* __restrict__ st, float* __restrict__ Out) {
  __shared__ float As[16 * LDS_STRIDE];
  __shared__ float s_sc[CH];
  __shared__ float s_tr[CH];
  const int tid = threadIdx.x;
  if (tid < CH) { s_sc[tid] = st[tid]; s_tr[tid] = st[CH + tid]; }
  __syncthreads();

  const long long row0 = (long long)blockIdx.x * 16;
  for (int i = tid * 4; i < 16 * CH; i += 256 * 4) {
    const int r = i >> 7, c = i & 127;
    float4 v = *(const float4*)&H[(row0 + r) * CH + c];
    v.x = fmaxf(fmaf(v.x, s_sc[c + 0], s_tr[c + 0]), 0.f);
    v.y = fmaxf(fmaf(v.y, s_sc[c + 1], s_tr[c + 1]), 0.f);
    v.z = fmaxf(fmaf(v.z, s_sc[c + 2], s_tr[c + 2]), 0.f);
    v.w = fmaxf(fmaf(v.w, s_sc[c + 3], s_tr[c + 3]), 0.f);
    *(float4*)&As[r * LDS_STRIDE + c] = v;
  }
  __syncthreads();

  const int lane  = tid & 31;
  const int mlane = lane & 15;
  const int kofs  = (lane >> 4) << 1;
  const int n0    = (tid >> 5) * 16;

  const float* arow = &As[mlane * LDS_STRIDE];
  const float* wrow = &W[(size_t)(n0 + mlane) * CH];

  v8f acc = {};
  #pragma unroll 4
  for (int k = 0; k < CH; k += 4) {
    v2f a, b;
    a.x = arow[k + kofs]; a.y = arow[k + kofs + 1];
    b.x = wrow[k + kofs]; b.y = wrow[k + kofs + 1];
    acc = wmma_fp32(a, b, acc);
  }

  const int col = n0 + mlane;
  const float bv = bias[col];
  const long long rbase = row0 + ((lane >> 4) << 3);
  #pragma unroll
  for (int v = 0; v < 8; ++v)
    Out[(rbase + v) * CH + col] = acc[v] + bv;
}

// ---------------------------------------------------------------------------
// Launch. Inputs: 0=x, 1=edge_index(i64), 2=eps, 3=W1, 4=b1, 5=gamma, 6=beta,
//                 7=W2, 8=b2. Output: [N,128] f32 in d_out.
// ws: [ buf: N*128 f32 | sums: 256 f32 | st: 256 f32 ]
// ---------------------------------------------------------------------------
extern "C" void kernel_launch(void* const* d_in, const int* in_sizes, int n_in,
                              void* d_out, int out_size, void* d_ws, size_t ws_size,
                              hipStream_t stream) {
  const float*     x     = (const float*)d_in[0];
  const long long* ei    = (const long long*)d_in[1];
  const float*     eps   = (const float*)d_in[2];
  const float*     W1    = (const float*)d_in[3];
  const float*     b1    = (const float*)d_in[4];
  const float*     gamma = (const float*)d_in[5];
  const float*     beta  = (const float*)d_in[6];
  const float*     W2    = (const float*)d_in[7];
  const float*     b2    = (const float*)d_in[8];

  const int N = in_sizes[0] / CH;       // 100000
  const int E = in_sizes[1] / 2;        // 1600000

  float* buf  = (float*)d_ws;           // N*CH aggregation buffer
  float* sums = buf + (size_t)N * CH;   // 256 floats
  float* st   = sums + 2 * CH;          // 256 floats
  float* H    = (float*)d_out;          // h lives in d_out; final GEMM overwrites in place

  // 1) buf = (1+eps)*x ; zero BN sums
  const int total4 = (N * CH) / 4;
  k_init<<<(total4 + 255) / 256, 256, 0, stream>>>(x, eps, buf, sums, total4);

  // 2) scatter-add edges (wave per edge, grid-stride)
  k_edge_scatter<<<4096, 256, 0, stream>>>(x, ei, buf, E);

  // 3) h = buf @ W1^T + b1   (N/16 row tiles)
  k_gemm_bias<<<N / 16, 256, 0, stream>>>(buf, W1, b1, H);

  // 4) per-channel batch stats
  k_bn_stats<<<1024, 256, 0, stream>>>(H, sums, N);

  // 5) fold BN -> scale/shift
  k_bn_finalize<<<1, CH, 0, stream>>>(sums, gamma, beta, st, 1.0f / (float)N);

  // 6) out = relu(h*s+t) @ W2^T + b2 (in place over H)
  k_bn_relu_gemm<<<N / 16, 256, 0, stream>>>(H, W2, b2, st, H);

  (void)n_in; (void)out_size; (void)ws_size;
}